// GNN_Local_71347996721325
// MI455X (gfx1250) — compile-verified
//
#include <hip/hip_runtime.h>
#include <hip/hip_bf16.h>

typedef __attribute__((ext_vector_type(2))) float v2f;
typedef __attribute__((ext_vector_type(8))) float v8f;

#define EPS_BN   1e-5f
#define SLOPE_LR 0.01f

// ---------------------------------------------------------------------------
// deg[n] += edge_w[e] for dst[e]==n
__global__ void deg_kernel(const int* __restrict__ dst,
                           const float* __restrict__ ew,
                           float* __restrict__ deg, int E) {
    int e = blockIdx.x * blockDim.x + threadIdx.x;
    if (e >= E) return;
    atomicAdd(deg + dst[e], ew[e]);
}

// w[e] = dis[src]*ew*dis[dst]
__global__ void edge_norm_kernel(const int* __restrict__ src,
                                 const int* __restrict__ dst,
                                 const float* __restrict__ ew,
                                 const float* __restrict__ deg,
                                 float* __restrict__ w, int E) {
    int e = blockIdx.x * blockDim.x + threadIdx.x;
    if (e >= E) return;
    float ds = deg[src[e]], dd = deg[dst[e]];
    float is = ds > 0.f ? rsqrtf(fmaxf(ds, 1e-30f)) : 0.f;
    float id = dd > 0.f ? rsqrtf(fmaxf(dd, 1e-30f)) : 0.f;
    w[e] = is * ew[e] * id;
}

// ---------------------------------------------------------------------------
// hout[b, dst[e], f..f+3] += h[b, src[e], f..f+3] * w[e]
// thread per (edge, batch, 4-feature chunk). hout must be pre-zeroed.
__global__ void prop_kernel(const float* __restrict__ h,
                            const int* __restrict__ src,
                            const int* __restrict__ dst,
                            const float* __restrict__ w,
                            float* __restrict__ hout,
                            int Nn, int E, int F, int B) {
    long long idx = (long long)blockIdx.x * blockDim.x + threadIdx.x;
    int f4 = F >> 2;
    long long total = (long long)E * B * f4;
    if (idx >= total) return;
    int c = (int)(idx % f4);
    long long t = idx / f4;
    int b = (int)(t % B);
    int e = (int)(t / B);
    int s = src[e], d = dst[e];
    float ww = w[e];
    const float4 v = *(const float4*)(h + ((long long)(b * Nn + s)) * F + c * 4);
    float* o = hout + ((long long)(b * Nn + d)) * F + c * 4;
    atomicAdd(o + 0, v.x * ww);
    atomicAdd(o + 1, v.y * ww);
    atomicAdd(o + 2, v.z * ww);
    atomicAdd(o + 3, v.w * ww);
}

// ---------------------------------------------------------------------------
// C[M,Ncol] (+)= A[M,K] @ W[K,Ncol] using V_WMMA_F32_16X16X4_F32.
// One wave per 16x16 output tile; K stepped by 4.
// A VGPR layout (16x4 f32): lanes 0-15 row M=lane, {K0,K1}; lanes 16-31 row M=lane-16, {K2,K3}
// B symmetric on columns; C/D: 8 VGPRs, lane<16 rows 0-7 / lane>=16 rows 8-15, col = lane&15.
__global__ void wmma_gemm_kernel(const float* __restrict__ A,
                                 const float* __restrict__ W,
                                 float* __restrict__ C,
                                 int M, int K, int Ncol, int accumulate) {
    int wave = (int)(((long long)blockIdx.x * blockDim.x + threadIdx.x) >> 5);
    int lane = threadIdx.x & 31;
    int colTiles = Ncol >> 4;
    int numTiles = (M >> 4) * colTiles;
    if (wave >= numTiles) return;   // wave-uniform: EXEC stays all-ones past here
    int tm = wave / colTiles;
    int tn = wave - tm * colTiles;
    int row0 = tm << 4;
    int col0 = tn << 4;
    int l15  = lane & 15;
    int kp   = (lane >> 4) << 1;        // 0 or 2
    int rbase = row0 + ((lane >> 4) << 3);
    int cidx  = col0 + l15;

    v8f c;
    if (accumulate) {
#pragma unroll
        for (int i = 0; i < 8; ++i) c[i] = C[(long long)(rbase + i) * Ncol + cidx];
    } else {
        c = (v8f){0.f, 0.f, 0.f, 0.f, 0.f, 0.f, 0.f, 0.f};
    }

    const float* arow = A + (long long)(row0 + l15) * K;
    for (int kb = 0; kb < K; kb += 4) {
        v2f a, b;
        a.x = arow[kb + kp];
        a.y = arow[kb + kp + 1];
        b.x = W[(long long)(kb + kp) * Ncol + cidx];
        b.y = W[(long long)(kb + kp + 1) * Ncol + cidx];
        c = __builtin_amdgcn_wmma_f32_16x16x4_f32(false, a, false, b,
                                                  (short)0, c, false, false);
    }
#pragma unroll
    for (int i = 0; i < 8; ++i) C[(long long)(rbase + i) * Ncol + cidx] = c[i];
}

// ---------------------------------------------------------------------------
// Per-node BatchNorm over (B, F) = 256 values + LeakyReLU.
// Block = 256 threads (b = t>>6, f = t&63), one block per node. F fixed at 64.
__global__ void bn_act_kernel(const float* __restrict__ acc,
                              const float* __restrict__ bias,
                              const float* __restrict__ g,
                              const float* __restrict__ be,
                              float* __restrict__ hout, int Nn) {
    __shared__ float ssum[256];
    __shared__ float ssq[256];
    int n = blockIdx.x;
    int t = threadIdx.x;
    int b = t >> 6, f = t & 63;
    long long idx = ((long long)(b * Nn + n)) * 64 + f;
    float v = acc[idx] + bias[f];
    ssum[t] = v;
    ssq[t]  = v * v;
    __syncthreads();
    for (int s = 128; s > 0; s >>= 1) {
        if (t < s) { ssum[t] += ssum[t + s]; ssq[t] += ssq[t + s]; }
        __syncthreads();
    }
    float mu  = ssum[0] * (1.0f / 256.0f);
    float var = ssq[0] * (1.0f / 256.0f) - mu * mu;   // biased, as torch normalizes
    float r   = rsqrtf(var + EPS_BN);
    float y   = g[n] * (v - mu) * r + be[n];
    y = y > 0.f ? y : SLOPE_LR * y;
    hout[idx] = y;
}

// ---------------------------------------------------------------------------
// out[M,2] (+)= h[M,K] @ W[K,2]  (layer-3 dense part; trivially small)
__global__ void gemm2_kernel(const float* __restrict__ h,
                             const float* __restrict__ W,
                             float* __restrict__ out,
                             int M, int K, int accumulate) {
    int r = blockIdx.x * blockDim.x + threadIdx.x;
    if (r >= M) return;
    float o0 = accumulate ? out[(long long)r * 2 + 0] : 0.f;
    float o1 = accumulate ? out[(long long)r * 2 + 1] : 0.f;
    const float* hp = h + (long long)r * K;
    for (int k = 0; k < K; ++k) {
        float v = hp[k];
        o0 += v * W[k * 2 + 0];
        o1 += v * W[k * 2 + 1];
    }
    out[(long long)r * 2 + 0] = o0;
    out[(long long)r * 2 + 1] = o1;
}

// ---------------------------------------------------------------------------
// Final: add b3, denorm (std with inf->0, mean, mask), per-node clamp.
__global__ void final_kernel(const float* __restrict__ acc3,
                             const float* __restrict__ b3,
                             const float* __restrict__ mask,
                             const float* __restrict__ mean_y,
                             const float* __restrict__ std_y,
                             const float* __restrict__ vmin,
                             const float* __restrict__ vmax,
                             const float* __restrict__ qmin,
                             const float* __restrict__ qmax,
                             float* __restrict__ out, int Nn, int M) {
    int r = blockIdx.x * blockDim.x + threadIdx.x;
    if (r >= M) return;
    int n = r % Nn;
    float v0 = acc3[(long long)r * 2 + 0] + b3[0];
    float v1 = acc3[(long long)r * 2 + 1] + b3[1];
    float s0 = std_y[n * 2 + 0]; if (isinf(s0)) s0 = 0.f;
    float s1 = std_y[n * 2 + 1]; if (isinf(s1)) s1 = 0.f;
    float o0 = (v0 * s0 + mean_y[n * 2 + 0]) * mask[n * 2 + 0];
    float o1 = (v1 * s1 + mean_y[n * 2 + 1]) * mask[n * 2 + 1];
    o0 = fminf(fmaxf(o0, vmin[n]), vmax[n]);
    o1 = fminf(fmaxf(o1, qmin[n]), qmax[n]);
    out[(long long)r * 2 + 0] = o0;
    out[(long long)r * 2 + 1] = o1;
}

// ---------------------------------------------------------------------------
static inline int cdiv(long long a, int b) { return (int)((a + b - 1) / b); }

extern "C" void kernel_launch(void* const* d_in, const int* in_sizes, int n_in,
                              void* d_out, int out_size, void* d_ws, size_t ws_size,
                              hipStream_t stream) {
    const float* x     = (const float*)d_in[0];
    const int*   src   = (const int*)d_in[1];
    const int*   dst   = (const int*)d_in[2];
    const float* ew    = (const float*)d_in[3];
    const float* W1    = (const float*)d_in[4];
    const float* b1    = (const float*)d_in[5];
    const float* W2    = (const float*)d_in[6];
    const float* b2    = (const float*)d_in[7];
    const float* W3    = (const float*)d_in[8];
    const float* b3    = (const float*)d_in[9];
    const float* g1    = (const float*)d_in[10];
    const float* be1   = (const float*)d_in[11];
    const float* g2    = (const float*)d_in[12];
    const float* be2   = (const float*)d_in[13];
    const float* mask  = (const float*)d_in[14];
    const float* meany = (const float*)d_in[15];
    const float* stdy  = (const float*)d_in[16];
    const float* vmin  = (const float*)d_in[17];
    const float* vmax  = (const float*)d_in[18];
    const float* qmin  = (const float*)d_in[19];
    const float* qmax  = (const float*)d_in[20];

    const int Nn = in_sizes[10];            // g1: [N]
    const int E  = in_sizes[1];             // src: [E]
    const int B  = in_sizes[0] / (Nn * 4);  // x: [B,N,4]
    const int M  = B * Nn;                  // GEMM rows
    const int F  = 64;

    // ---- carve workspace (floats), 256-byte aligned chunks ----
    float* ws = (float*)d_ws;
    size_t off = 0;
    auto carve = [&](size_t nfloats) {
        float* p = ws + off;
        off += (nfloats + 63) & ~(size_t)63;
        return p;
    };
    float* deg  = carve((size_t)Nn);
    float* wn   = carve((size_t)E);
    float* bufA = carve((size_t)M * F);
    float* bufB = carve((size_t)M * F);
    float* acc  = carve((size_t)M * F);
    float* acc3 = carve((size_t)M * 2);
    (void)ws_size;

    // ---- edge normalization ----
    hipMemsetAsync(deg, 0, (size_t)Nn * sizeof(float), stream);
    deg_kernel<<<cdiv(E, 256), 256, 0, stream>>>(dst, ew, deg, E);
    edge_norm_kernel<<<cdiv(E, 256), 256, 0, stream>>>(src, dst, ew, deg, wn, E);

    auto prop = [&](const float* hin, float* hout, int Fin) {
        hipMemsetAsync(hout, 0, (size_t)M * Fin * sizeof(float), stream);
        long long tot = (long long)E * B * (Fin >> 2);
        prop_kernel<<<cdiv(tot, 256), 256, 0, stream>>>(hin, src, dst, wn, hout,
                                                        Nn, E, Fin, B);
    };
    auto gemm = [&](const float* A, const float* W, float* C, int K, int Ncol, int accum) {
        long long waves = (long long)(M >> 4) * (Ncol >> 4);
        gemm2_kernel; // (silence unused warning path; no-op expression)
        wmma_gemm_kernel<<<cdiv(waves * 32, 256), 256, 0, stream>>>(A, W, C, M, K, Ncol, accum);
    };

    // ================= Layer 1: 4 -> 64 =================
    gemm(x, W1 + 0 * 4 * 64, acc, 4, 64, 0);
    prop(x, bufB, 4);
    gemm(bufB, W1 + 1 * 4 * 64, acc, 4, 64, 1);
    prop(bufB, bufA, 4);
    gemm(bufA, W1 + 2 * 4 * 64, acc, 4, 64, 1);
    prop(bufA, bufB, 4);
    gemm(bufB, W1 + 3 * 4 * 64, acc, 4, 64, 1);
    bn_act_kernel<<<Nn, 256, 0, stream>>>(acc, b1, g1, be1, bufA, Nn); // h1 -> bufA

    // ================= Layer 2: 64 -> 64 =================
    gemm(bufA, W2 + 0 * 64 * 64, acc, 64, 64, 0);
    prop(bufA, bufB, 64);
    gemm(bufB, W2 + 1 * 64 * 64, acc, 64, 64, 1);
    prop(bufB, bufA, 64);                     // bufA (old h) no longer needed
    gemm(bufA, W2 + 2 * 64 * 64, acc, 64, 64, 1);
    prop(bufA, bufB, 64);
    gemm(bufB, W2 + 3 * 64 * 64, acc, 64, 64, 1);
    bn_act_kernel<<<Nn, 256, 0, stream>>>(acc, b2, g2, be2, bufA, Nn); // h2 -> bufA

    // ================= Layer 3: 64 -> 2 =================
    gemm2_kernel<<<cdiv(M, 256), 256, 0, stream>>>(bufA, W3 + 0 * 64 * 2, acc3, M, 64, 0);
    prop(bufA, bufB, 64);
    gemm2_kernel<<<cdiv(M, 256), 256, 0, stream>>>(bufB, W3 + 1 * 64 * 2, acc3, M, 64, 1);
    prop(bufB, bufA, 64);
    gemm2_kernel<<<cdiv(M, 256), 256, 0, stream>>>(bufA, W3 + 2 * 64 * 2, acc3, M, 64, 1);
    prop(bufA, bufB, 64);
    gemm2_kernel<<<cdiv(M, 256), 256, 0, stream>>>(bufB, W3 + 3 * 64 * 2, acc3, M, 64, 1);

    // ================= Denorm + clamp =================
    final_kernel<<<cdiv(M, 256), 256, 0, stream>>>(acc3, b3, mask, meany, stdy,
                                                   vmin, vmax, qmin, qmax,
                                                   (float*)d_out, Nn, M);
}